// HybridFCL_65481071401225
// MI455X (gfx1250) — compile-verified
//
#include <hip/hip_runtime.h>

// QLSTM recurrence for MI455X (gfx1250, wave32), round 4.
// Per wave: one 16-row tile of T=1024. Recurrent step:
//   acc = hx(16x4) x B1 + acc0[s]        [V_WMMA_F32_16X16X4_F32, critical path]
//   acc0[s+1] = feats(16x4) x B0 + (b+theta)/2pi   [pipelined off the path]
// v_cos_f32 directly on WMMA output (B/bias pre-scaled by 1/2pi), nonlinearities
// via CDNA5 V_TANH_F32, qubit products 2 bpermutes, gate combine 2 bpermutes,
// D->A transpose through 256B wave-private LDS, coalesced 2x128B output stores.

typedef float v2f __attribute__((ext_vector_type(2)));
typedef float v8f __attribute__((ext_vector_type(8)));

#define NSTEPS 256   // B (scan axis)
#define TLEN   1024  // T
#define DDIM   64    // D
#define PROTON 64    // P
#define INV2PI 0.15915494309189535f

__device__ __forceinline__ float fast_tanh(float x) {
#if __has_builtin(__builtin_amdgcn_tanhf)
    return __builtin_amdgcn_tanhf(x);          // CDNA5 V_TANH_F32
#elif __has_builtin(__builtin_amdgcn_tanh_f32)
    return __builtin_amdgcn_tanh_f32(x);
#else
    const float e = __expf(-2.f * x);
    return 2.f / (1.f + e) - 1.f;
#endif
}

__global__ __launch_bounds__(32) void qlstm_wmma_kernel(
    const float* __restrict__ x,     const float* __restrict__ proto,
    const float* __restrict__ W_lin, const float* __restrict__ b_lin,
    const float* __restrict__ W_f, const float* __restrict__ b_f, const float* __restrict__ th_f,
    const float* __restrict__ W_i, const float* __restrict__ b_i, const float* __restrict__ th_i,
    const float* __restrict__ W_u, const float* __restrict__ b_u, const float* __restrict__ th_u,
    const float* __restrict__ W_o, const float* __restrict__ b_o, const float* __restrict__ th_o,
    float* __restrict__ out)
{
    __shared__ __align__(16) float hxbuf[64];   // [h][t_local] = 4 x 16, wave-private

    const int lane = threadIdx.x & 31;
    const int half = lane >> 4;            // 0/1
    const int n    = lane & 15;            // N column of B/C/D; M row of A
    const int q    = lane & 3;             // qubit index within gate
    const int gate = (lane >> 2) & 3;      // 0=f 1=i 2=u 3=o
    const int tile = blockIdx.x;           // 64 tiles of 16 rows
    const int tA   = tile * 16 + n;        // A-matrix row -> global t

    // per-lane gate pointers (uniform per 4-lane group)
    const float* Wg = (gate == 0) ? W_f  : (gate == 1) ? W_i  : (gate == 2) ? W_u  : W_o;
    const float* bg = (gate == 0) ? b_f  : (gate == 1) ? b_i  : (gate == 2) ? b_u  : b_o;
    const float* tg = (gate == 0) ? th_f : (gate == 1) ? th_i : (gate == 2) ? th_u : th_o;

    const int k0 = 2 * half;               // K rows held by this lane (A & B layout)
    const int k1 = 2 * half + 1;

    // B operands pre-scaled by 1/(2pi): WMMA output directly in v_cos units.
    v2f b0v, b1v;
    b0v.x = INV2PI * Wg[(k0)     * 4 + q];
    b0v.y = INV2PI * Wg[(k1)     * 4 + q];
    b1v.x = INV2PI * Wg[(k0 + 4) * 4 + q];
    b1v.y = INV2PI * Wg[(k1 + 4) * 4 + q];

    // fold gate bias + theta (pre-scaled) into the accumulator
    const float cb = INV2PI * (bg[q] + tg[q]);
    v8f cbias;
    #pragma unroll
    for (int v = 0; v < 8; ++v) cbias[v] = cb;

    // feats[s,t,h] = cos(x[s,t,0]) * vsum[h] + b_lin[h];  vsum = (cos proto[:,0]) @ W_lin
    float vs0 = 0.f, vs1 = 0.f;
    for (int p = 0; p < PROTON; ++p) {
        float cp = __cosf(proto[p * 4]);
        vs0 += cp * W_lin[p * 4 + k0];
        vs1 += cp * W_lin[p * 4 + k1];
    }
    const float bl0 = b_lin[k0];
    const float bl1 = b_lin[k1];

    // ---- lane-constant branchless select state (hoisted compares) ----
    const bool q_odd   = (q & 1) != 0;
    const bool q_zero  = (q == 0);
    const bool skip_op = (q == 1);
    const float nl_scale = (gate == 2) ? 1.f : 0.5f;
    const float nl_amp   = (gate == 2) ? 1.f : 0.5f;
    const float nl_off   = (gate == 2) ? 0.f : 0.5f;
    const int r = lane >> 2;               // 0..7: row index for coalesced output

    // cell state (valid in gate-0 lanes; others hold don't-care)
    float cstate[8];
    #pragma unroll
    for (int v = 0; v < 8; ++v) cstate[v] = 0.f;
    if (lane < 16) {                       // zero hx for step 0
        hxbuf[lane] = 0.f; hxbuf[lane + 16] = 0.f;
        hxbuf[lane + 32] = 0.f; hxbuf[lane + 48] = 0.f;
    }

    // ---- pipeline preamble: acc0 for s=0, prefetch x[1] ----
    v8f acc0;
    {
        const float cx = __cosf(x[(size_t)tA * DDIM]);
        v2f a0; a0.x = __builtin_fmaf(cx, vs0, bl0); a0.y = __builtin_fmaf(cx, vs1, bl1);
        acc0 = __builtin_amdgcn_wmma_f32_16x16x4_f32(
            false, a0, false, b0v, (short)0, cbias, false, false);
    }
    float xv = x[((size_t)1 * TLEN + tA) * DDIM];

    for (int s = 0; s < NSTEPS; ++s) {
        // recurrent WMMA (critical path): hx tile from LDS
        v2f a1; a1.x = hxbuf[k0 * 16 + n]; a1.y = hxbuf[k1 * 16 + n];
        v8f acc = __builtin_amdgcn_wmma_f32_16x16x4_f32(
            false, a1, false, b1v, (short)0, acc0, false, false);

        // pipelined: acc0 for step s+1 from prefetched xv = x[s+1] (off the path)
        {
            const float cx = __cosf(xv);
            v2f a0; a0.x = __builtin_fmaf(cx, vs0, bl0); a0.y = __builtin_fmaf(cx, vs1, bl1);
            acc0 = __builtin_amdgcn_wmma_f32_16x16x4_f32(
                false, a0, false, b0v, (short)0, cbias, false, false);
        }
        // prefetch x[s+2] (clamped, no branch)
        const int sn = (s < NSTEPS - 2) ? s + 2 : NSTEPS - 1;
        xv = x[((size_t)sn * TLEN + tA) * DDIM];

        float hn[8];
        #pragma unroll
        for (int v = 0; v < 8; ++v) {
            const float a  = __builtin_amdgcn_cosf(acc[v]);   // angle in revolutions
            const float p1 = __shfl_xor(a, 1);
            const float pp = a * p1;                  // own pair product
            const float op = __shfl_xor(pp, 2);       // other pair product
            const float t1 = q_odd ? pp : (q_zero ? p1 : a);
            const float t2 = skip_op ? 1.f : op;
            const float z  = t1 * t2;
            const float val = __builtin_fmaf(nl_amp, fast_tanh(nl_scale * z), nl_off);
            // hierarchical gate combine: s12 swaps f<->o,i<->g; m=i*g at group1;
            // mg = xor4(m) delivers i*g to group0 (which owns the writeback).
            const float s12 = __shfl_xor(val, 12);
            const float m   = val * s12;
            const float mg  = __shfl_xor(m, 4);
            const float cn = __builtin_fmaf(val, cstate[v], mg);  // f*c + i*g (group0)
            cstate[v] = cn;
            hn[v] = s12 * fast_tanh(cn);                          // o * tanh(c') (group0)
        }

        // single tiny divergent region: D-layout -> A-layout transpose (2x b128)
        if (gate == 0) {
            float4 lo = make_float4(hn[0], hn[1], hn[2], hn[3]);
            float4 hi = make_float4(hn[4], hn[5], hn[6], hn[7]);
            *(float4*)&hxbuf[q * 16 + 8 * half]     = lo;   // rows +0..3
            *(float4*)&hxbuf[q * 16 + 8 * half + 4] = hi;   // rows +4..7
        }

        // coalesced output: 64 contiguous floats per tile per step (2x128B/wave)
        const float o0 = hxbuf[q * 16 + r];        // (t=r,   h=q) -> flat idx = lane
        const float o1 = hxbuf[q * 16 + r + 8];    // (t=r+8, h=q) -> flat idx = lane+32
        const size_t ob = (size_t)s * (TLEN * 4) + (size_t)tile * 64;
        out[ob + lane]      = o0;
        out[ob + lane + 32] = o1;
    }
}

extern "C" void kernel_launch(void* const* d_in, const int* in_sizes, int n_in,
                              void* d_out, int out_size, void* d_ws, size_t ws_size,
                              hipStream_t stream) {
    const float* x      = (const float*)d_in[0];
    const float* proto  = (const float*)d_in[1];
    const float* W_lin  = (const float*)d_in[2];
    const float* b_lin  = (const float*)d_in[3];
    const float* W_f    = (const float*)d_in[4];
    const float* b_f    = (const float*)d_in[5];
    const float* th_f   = (const float*)d_in[6];
    const float* W_i    = (const float*)d_in[7];
    const float* b_i    = (const float*)d_in[8];
    const float* th_i   = (const float*)d_in[9];
    const float* W_u    = (const float*)d_in[10];
    const float* b_u    = (const float*)d_in[11];
    const float* th_u   = (const float*)d_in[12];
    const float* W_o    = (const float*)d_in[13];
    const float* b_o    = (const float*)d_in[14];
    const float* th_o   = (const float*)d_in[15];
    float* out = (float*)d_out;

    dim3 grid(TLEN / 16);   // 64 single-wave workgroups, one 16-row tile each
    dim3 block(32);
    qlstm_wmma_kernel<<<grid, block, 0, stream>>>(
        x, proto, W_lin, b_lin,
        W_f, b_f, th_f, W_i, b_i, th_i, W_u, b_u, th_u, W_o, b_o, th_o,
        out);
}